// ILloss_83794811945270
// MI455X (gfx1250) — compile-verified
//
#include <hip/hip_runtime.h>

// ---------------------------------------------------------------------------
// ILloss on MI455X (gfx1250), compile-only tuned.
//
// Key algebraic collapse:  sum_{i!=j} cos(c_i,c_j) = ||sum_i c_i/|c_i|||^2 - K
// so the 8192x8192x64 GEMM becomes an O(K*D) reduction. Total HBM traffic
// ~10 MB -> ~0.4us at 23.3 TB/s; everything is bandwidth bound.
// The column-sum reduction u = 1^T @ N is done with V_WMMA_F32_16X16X4_F32
// (all-ones A matrix => each C row = column sums of the 4x16 B tile), which
// keeps the reduction on the CDNA5 matrix path at no extra memory cost.
// ---------------------------------------------------------------------------

typedef __attribute__((ext_vector_type(2))) float v2f;
typedef __attribute__((ext_vector_type(8))) float v8f;

#define NUM_CLASSES 8192
#define FEAT_DIM    64
#define BATCH       16384
#define LAMB        0.01
#define LAMB1       10.0

// ws layout (floats): [0] = sum of ||feat - centers[y]||^2
//                     [1..64]   = u accumulator (64)
//                     [65..65+8191] = rnorm per class
__global__ void zero_ws_kernel(float* __restrict__ ws) {
    int t = threadIdx.x;
    if (t < 65) ws[t] = 0.0f;
}

// ---- center loss: 16 float4-lanes per row, fully coalesced ---------------
__global__ __launch_bounds__(256)
void center_loss_kernel(const int* __restrict__ y,
                        const float4* __restrict__ feat4,
                        const float4* __restrict__ cent4,
                        float* __restrict__ loss_acc) {
    int g   = blockIdx.x * 256 + threadIdx.x;   // BATCH*16 threads
    int row = g >> 4;
    int v   = g & 15;
    int cls = y[row];
    float4 f = feat4[g];
    float4 c = cent4[cls * 16 + v];
    float dx = f.x - c.x, dy = f.y - c.y, dz = f.z - c.z, dw = f.w - c.w;
    float acc = dx * dx + dy * dy + dz * dz + dw * dw;

    // wave32 reduce
    #pragma unroll
    for (int off = 16; off > 0; off >>= 1) acc += __shfl_down(acc, off, 32);

    __shared__ float sm[8];
    int lane = threadIdx.x & 31, wid = threadIdx.x >> 5;
    if (lane == 0) sm[wid] = acc;
    __syncthreads();
    if (wid == 0) {
        float a = (lane < 8) ? sm[lane] : 0.0f;
        #pragma unroll
        for (int off = 4; off > 0; off >>= 1) a += __shfl_down(a, off, 32);
        if (lane == 0) atomicAdd(loss_acc, a);
    }
}

// ---- per-class inverse norm: 16 lanes per row, coalesced float4 ----------
__global__ __launch_bounds__(256)
void rnorm_kernel(const float4* __restrict__ cent4, float* __restrict__ rnorm) {
    int g = blockIdx.x * 256 + threadIdx.x;     // NUM_CLASSES*16 threads
    float4 c = cent4[g];
    float s = c.x * c.x + c.y * c.y + c.z * c.z + c.w * c.w;
    // reduce within each 16-lane half-wave (xor masks stay inside the half)
    s += __shfl_xor(s, 1, 32);
    s += __shfl_xor(s, 2, 32);
    s += __shfl_xor(s, 4, 32);
    s += __shfl_xor(s, 8, 32);
    if ((g & 15) == 0) rnorm[g >> 4] = rsqrtf(s);
}

// ---- u = columnwise sum of row-normalized centers via WMMA ---------------
// Each wave owns one 16-column group (cg) and 32 rows; per iteration it feeds
// a 4x16 tile of N = diag(rnorm)*centers as B, with A = all-ones 16x4, so
// C[m][n] += sum_k B[k][n]. With A==1 the k-slot permutation inside B's lane
// layout is irrelevant; only n = lane&15 matters (matches C/D layout).
__global__ __launch_bounds__(256)
void usum_wmma_kernel(const float* __restrict__ centers,
                      const float* __restrict__ rnorm,
                      float* __restrict__ u) {
    int tid  = threadIdx.x;
    int gwid = (blockIdx.x * 256 + tid) >> 5;   // global wave id (1024 total)
    int lane = tid & 31;
    int cg   = gwid & 3;                        // column group: cols cg*16..+15
    int part = gwid >> 2;                       // row partition: 0..255
    int n    = lane & 15;
    int h    = lane >> 4;
    int col  = cg * 16 + n;

    v2f a = {1.0f, 1.0f};                       // all-ones A (16x4)
    v8f c = {};                                 // accumulator

    int rbase = part * 32;                      // 32 rows per wave, 4 per WMMA
    #pragma unroll
    for (int q = 0; q < 8; ++q) {
        int r  = rbase + q * 4;
        int r0 = r + h;
        int r1 = r + 2 + h;
        v2f b;
        b[0] = centers[r0 * FEAT_DIM + col] * rnorm[r0];
        b[1] = centers[r1 * FEAT_DIM + col] * rnorm[r1];
        c = __builtin_amdgcn_wmma_f32_16x16x4_f32(
                /*neg_a=*/false, a, /*neg_b=*/false, b,
                /*c_mod=*/(short)0, c, /*reuse_a=*/false, /*reuse_b=*/false);
    }
    // every C row is identical; c[0] on lanes 0..15 holds col sums for n=lane
    if (lane < 16) atomicAdd(&u[cg * 16 + lane], c[0]);
}

// ---- finalize: out = LAMB*(loss_cen + LAMB1*(||u||^2 - K + K(K-1))) ------
__global__ void finalize_kernel(const float* __restrict__ ws,
                                float* __restrict__ out) {
    int t = threadIdx.x;                        // 64 threads
    float uv = ws[1 + t];
    float s  = uv * uv;
    #pragma unroll
    for (int off = 16; off > 0; off >>= 1) s += __shfl_down(s, off, 32);
    __shared__ float sm[2];
    if ((t & 31) == 0) sm[t >> 5] = s;
    __syncthreads();
    if (t == 0) {
        double usq = (double)sm[0] + (double)sm[1];
        double K   = (double)NUM_CLASSES;
        double la  = usq - K + K * (K - 1.0);
        double lc  = (double)ws[0] * 0.5 / (double)BATCH;
        out[0] = (float)(LAMB * (lc + LAMB1 * la));
    }
}

extern "C" void kernel_launch(void* const* d_in, const int* in_sizes, int n_in,
                              void* d_out, int out_size, void* d_ws, size_t ws_size,
                              hipStream_t stream) {
    const int*   y       = (const int*)d_in[0];
    const float* feat    = (const float*)d_in[1];
    const float* centers = (const float*)d_in[2];
    float* out = (float*)d_out;
    float* ws  = (float*)d_ws;
    float* u     = ws + 1;
    float* rnorm = ws + 65;

    zero_ws_kernel<<<1, 96, 0, stream>>>(ws);
    center_loss_kernel<<<(BATCH * 16) / 256, 256, 0, stream>>>(
        y, (const float4*)feat, (const float4*)centers, ws);
    rnorm_kernel<<<(NUM_CLASSES * 16) / 256, 256, 0, stream>>>(
        (const float4*)centers, rnorm);
    usum_wmma_kernel<<<128, 256, 0, stream>>>(centers, rnorm, u);
    finalize_kernel<<<1, 64, 0, stream>>>(ws, out);
}